// model_86397562126851
// MI455X (gfx1250) — compile-verified
//
#include <hip/hip_runtime.h>
#include <stdint.h>

// Pairwise box IoU + union, N=M=4096, f32.
// Store-bound: 134 MB of output -> ~5.8us floor at 23.3 TB/s HBM.
// CDNA5 paths: async global->LDS staging (ASYNCcnt + s_wait_asynccnt),
// ds_load_b128 tile reads, uniform s_load row boxes, NT b128 stores.
// 8-row blocking keeps y boxes in VGPRs and VALU (~3us) under the store floor:
// fast reciprocal (v_rcp_f32 + 1 Newton) instead of the ~10-op IEEE div chain.

typedef __attribute__((ext_vector_type(4))) float v4f;

#define ROWS_PER_BLOCK 8

__global__ __launch_bounds__(256) void iou_union_kernel(
    const float* __restrict__ x, const float* __restrict__ y,
    float* __restrict__ out_iou, float* __restrict__ out_uni, int M)
{
    __shared__ v4f ytile[1024];               // 16 KB: 1024 y-boxes per block
    const int tid   = threadIdx.x;
    const int i0    = blockIdx.y * ROWS_PER_BLOCK;   // first row of this block
    const int jbase = blockIdx.x << 10;              // first column of this block

    // ---- Stage y[jbase .. jbase+1023] into LDS with the async copy engine.
    // Thread t copies boxes {t, t+256, t+512, t+768}: coalesced 16B/lane.
    {
        const uint64_t gbase = (uint64_t)(uintptr_t)(y) + (uint64_t)jbase * 16ull;
        #pragma unroll
        for (int k = 0; k < 4; ++k) {
            const unsigned idx     = (unsigned)tid + (unsigned)(k << 8);
            const unsigned lds_off = (unsigned)(uintptr_t)(&ytile[idx]);
            const uint64_t gaddr   = gbase + (uint64_t)idx * 16ull;
            asm volatile("global_load_async_to_lds_b128 %0, %1, off"
                         :: "v"(lds_off), "v"(gaddr) : "memory");
        }
    }

    // ---- Row boxes: uniform per block -> scalar load path.
    v4f   xb[ROWS_PER_BLOCK];
    float area1[ROWS_PER_BLOCK];
    #pragma unroll
    for (int r = 0; r < ROWS_PER_BLOCK; ++r) {
        xb[r]    = ((const v4f*)x)[i0 + r];
        area1[r] = (xb[r].z - xb[r].x) * (xb[r].w - xb[r].y);
    }

    // Drain this wave's async copies, then barrier so all waves see the tile.
    asm volatile("s_wait_asynccnt 0x0" ::: "memory");
    __syncthreads();

    // ---- This thread's 4 y-boxes: read once from LDS, reused for 8 rows.
    v4f   yb[4];
    float area2[4];
    #pragma unroll
    for (int k = 0; k < 4; ++k) {
        yb[k]    = ytile[(tid << 2) + k];          // ds_load_b128
        area2[k] = (yb[k].z - yb[k].x) * (yb[k].w - yb[k].y);
    }

    const size_t colbase = (size_t)(jbase + (tid << 2));
    #pragma unroll
    for (int r = 0; r < ROWS_PER_BLOCK; ++r) {
        float ri[4], ru[4];
        #pragma unroll
        for (int k = 0; k < 4; ++k) {
            const float lx = fmaxf(xb[r].x, yb[k].x);
            const float ly = fmaxf(xb[r].y, yb[k].y);
            const float rx = fminf(xb[r].z, yb[k].z);
            const float ry = fminf(xb[r].w, yb[k].w);
            const float w  = fmaxf(rx - lx, 0.0f);
            const float h  = fmaxf(ry - ly, 0.0f);
            const float inter = w * h;
            const float uni   = (area1[r] + area2[k]) - inter;
            // Fast reciprocal: v_rcp_f32 + one Newton step (~1 ulp).
            float rcp = __builtin_amdgcn_rcpf(uni);
            rcp = __builtin_fmaf(__builtin_fmaf(-uni, rcp, 1.0f), rcp, rcp);
            ri[k] = inter * rcp;
            ru[k] = uni;
        }
        const size_t base = (size_t)(i0 + r) * (size_t)M + colbase;
        const v4f vi = { ri[0], ri[1], ri[2], ri[3] };
        const v4f vu = { ru[0], ru[1], ru[2], ru[3] };
        __builtin_nontemporal_store(vi, (v4f*)(out_iou + base));
        __builtin_nontemporal_store(vu, (v4f*)(out_uni + base));
    }
}

extern "C" void kernel_launch(void* const* d_in, const int* in_sizes, int n_in,
                              void* d_out, int out_size, void* d_ws, size_t ws_size,
                              hipStream_t stream) {
    const float* x = (const float*)d_in[0];   // [N,4] xyxy
    const float* y = (const float*)d_in[1];   // [M,4] xyxy
    const int N = in_sizes[0] / 4;
    const int M = in_sizes[1] / 4;

    float* out_iou = (float*)d_out;
    float* out_uni = out_iou + (size_t)N * (size_t)M;

    dim3 grid((unsigned)(M >> 10), (unsigned)(N / ROWS_PER_BLOCK));
    iou_union_kernel<<<grid, 256, 0, stream>>>(x, y, out_iou, out_uni, M);
}